// Model_35811437314119
// MI455X (gfx1250) — compile-verified
//
#include <hip/hip_runtime.h>
#include <hip/hip_bf16.h>
#include <math.h>

typedef float v2f __attribute__((ext_vector_type(2)));
typedef float v8f __attribute__((ext_vector_type(8)));

// ---------------------------------------------------------------------------
// Stage 1: LBS skinning. Per (bn, 16-point tile):
//   D(16x12) = W(16xJ) @ RT(Jx12)   via V_WMMA_F32_16X16X4_F32, K-chunks of 4.
// Operands staged in LDS (zero-padded) so the K-loop is branch-free:
// each fragment = one ds_load_b64. JC>0 gives a compile-time fully-unrolled loop.
// ---------------------------------------------------------------------------
#define LDS_STRIDE 68   // floats; 272B rows: v2f-aligned, banks (4m+j) -> conflict-free

template <int JC>
__global__ void skin_wmma_kernel(const float* __restrict__ xyz,
                                 const float* __restrict__ w,
                                 const float* __restrict__ Rg,
                                 const float* __restrict__ Tg,
                                 float* __restrict__ posed,
                                 int N, int NP, int Jrt) {
    const int J       = (JC > 0) ? JC : Jrt;
    const int nchunks = (J + 3) >> 2;
    const int JP      = nchunks * 4;          // padded K (<= 64)

    const int bn   = blockIdx.y;
    const int bx   = bn / N;                  // xyz / lbs_weights batch (broadcast over N)
    const int row0 = blockIdx.x * 16;
    const int lane = threadIdx.x;             // wave32: 0..31
    const int m    = lane & 15;
    const int half = lane >> 4;

    __shared__ float wld[16][LDS_STRIDE];     // A tile: weights [row][j]
    __shared__ float rtld[16][LDS_STRIDE];    // B tile transposed: [col][j]
    __shared__ float dtile[16][17];           // D tile (padded vs bank conflicts)

    // ---- stage A operand: 16 weight rows, zero-padded ----
    for (int r = 0; r < 16; ++r) {
        const int p = row0 + r;
        const float* wr = w + ((size_t)bx * NP + (p < NP ? p : 0)) * J;
        for (int j = lane; j < JP; j += 32)
            wld[r][j] = (p < NP && j < J) ? wr[j] : 0.0f;
    }
    // ---- stage B operand: RT = [R(j,9) | T(j,3)] transposed to [col][j] ----
    for (int j = lane; j < JP; j += 32) {
        const bool   jv   = (j < J);
        const size_t base = (size_t)bn * J + (jv ? j : 0);
        #pragma unroll
        for (int c = 0; c < 9; ++c)  rtld[c][j]     = jv ? Rg[base * 9 + c] : 0.0f;
        #pragma unroll
        for (int c = 0; c < 3; ++c)  rtld[9 + c][j] = jv ? Tg[base * 3 + c] : 0.0f;
        #pragma unroll
        for (int c = 12; c < 16; ++c) rtld[c][j]    = 0.0f;
    }
    __syncthreads();

    // ---- K loop: branch-free, 2x ds_load_b64 + 1x wmma per chunk ----
    v8f acc = {0.f, 0.f, 0.f, 0.f, 0.f, 0.f, 0.f, 0.f};
    #pragma unroll
    for (int c = 0; c < ((JC > 0) ? ((JC + 3) >> 2) : 1) * ((JC > 0) ? 1 : 0); ++c) { }
    if (JC > 0) {
        #pragma unroll
        for (int c = 0; c < ((JC + 3) >> 2); ++c) {
            const int j0 = c * 4 + half * 2;
            // A[m][k]: lane holds row m, k = (lane/16)*2 + {0,1}
            const v2f a = *(const v2f*)&wld[m][j0];
            // B[k][n]: lane holds col n=m, k = (lane/16)*2 + {0,1}
            const v2f b = *(const v2f*)&rtld[m][j0];
            acc = __builtin_amdgcn_wmma_f32_16x16x4_f32(
                false, a, false, b, (short)0, acc, false, false);
        }
    } else {
        for (int c = 0; c < nchunks; ++c) {
            const int j0 = c * 4 + half * 2;
            const v2f a = *(const v2f*)&wld[m][j0];
            const v2f b = *(const v2f*)&rtld[m][j0];
            acc = __builtin_amdgcn_wmma_f32_16x16x4_f32(
                false, a, false, b, (short)0, acc, false, false);
        }
    }

    // D layout: VGPR r, lane l -> D[(l/16)*8 + r][l%16]
    #pragma unroll
    for (int r = 0; r < 8; ++r)
        dtile[half * 8 + r][m] = acc[r];
    __syncthreads();

    // ---- finish: posed[p] = M(3x3) @ x[p] + T ----
    if (lane < 16) {
        const int pp = row0 + lane;
        if (pp < NP) {
            float Mrow[12];
            #pragma unroll
            for (int c = 0; c < 12; ++c) Mrow[c] = dtile[lane][c];
            const float x0 = xyz[((size_t)bx * NP + pp) * 3 + 0];
            const float x1 = xyz[((size_t)bx * NP + pp) * 3 + 1];
            const float x2 = xyz[((size_t)bx * NP + pp) * 3 + 2];
            #pragma unroll
            for (int k = 0; k < 3; ++k) {
                posed[((size_t)bn * NP + pp) * 3 + k] =
                    Mrow[3 * k + 0] * x0 + Mrow[3 * k + 1] * x1 +
                    Mrow[3 * k + 2] * x2 + Mrow[9 + k];
            }
        }
    }
}

// ---------------------------------------------------------------------------
// Stage 2: skel_hi[bn,i,:] = sum_t iw[i,t] * posed[bn, ip[i,t], :]
// ---------------------------------------------------------------------------
__global__ void interp_kernel(const float* __restrict__ posed,
                              const float* __restrict__ iw,
                              const int* __restrict__ ip,
                              float* __restrict__ skel,
                              int NP, int NPH, int total) {
    const int idx = blockIdx.x * blockDim.x + threadIdx.x;
    if (idx >= total) return;
    const int bn = idx / NPH;
    const int i  = idx - bn * NPH;

    float acc0 = 0.f, acc1 = 0.f, acc2 = 0.f;
    #pragma unroll
    for (int t = 0; t < 3; ++t) {
        const float wgt = iw[(size_t)i * 3 + t];
        const int   pt  = ip[(size_t)i * 3 + t];
        const float* src = posed + ((size_t)bn * NP + pt) * 3;
        acc0 += wgt * src[0];
        acc1 += wgt * src[1];
        acc2 += wgt * src[2];
    }
    float* dst = skel + ((size_t)bn * NPH + i) * 3;
    dst[0] = acc0; dst[1] = acc1; dst[2] = acc2;
}

// ---------------------------------------------------------------------------
// Stage 3: per (bn, face): Steiner frame, so3_exp, covariance transform.
// ---------------------------------------------------------------------------
__global__ void face_kernel(const float* __restrict__ skel,
                            const float* __restrict__ so3,
                            const float* __restrict__ scale,
                            const float* __restrict__ offset,
                            const int* __restrict__ faces,
                            float* __restrict__ out,
                            int N, int NPH, int NF, int total) {
    const int idx = blockIdx.x * blockDim.x + threadIdx.x;
    if (idx >= total) return;
    const int bn = idx / NF;
    const int f  = idx - bn * NF;
    const int b  = bn / N;                // so3/scale/offset batch (broadcast over N)

    float v[3][3];
    #pragma unroll
    for (int t = 0; t < 3; ++t) {
        const int vi = faces[(size_t)f * 3 + t];
        const float* s = skel + ((size_t)bn * NPH + vi) * 3;
        v[t][0] = s[0]; v[t][1] = s[1]; v[t][2] = s[2];
    }
    float cen[3];
    #pragma unroll
    for (int k = 0; k < 3; ++k)
        cen[k] = (v[0][k] + v[1][k] + v[2][k]) * (1.0f / 3.0f);

    const float c2 = 1.0f / (2.0f * 1.7320508075688772f); // 1/(2*sqrt(3))
    float f1[3], f2[3];
    #pragma unroll
    for (int k = 0; k < 3; ++k) {
        f1[k] = 0.5f * (v[2][k] - cen[k]);
        f2[k] = c2 * (v[1][k] - v[0][k]);
    }
    const float d11 = f1[0]*f1[0] + f1[1]*f1[1] + f1[2]*f1[2];
    const float d22 = f2[0]*f2[0] + f2[1]*f2[1] + f2[2]*f2[2];
    const float d12 = f1[0]*f2[0] + f1[1]*f2[1] + f1[2]*f2[2];
    const float t0  = 0.5f * atan2f(2.0f * d12, d11 - d22);
    const float ct = cosf(t0), st = sinf(t0);
    float a0[3], a1[3];
    #pragma unroll
    for (int k = 0; k < 3; ++k) {
        a0[k] =  f1[k] * ct + f2[k] * st;
        a1[k] = -f1[k] * st + f2[k] * ct;   // cos(t0+pi/2)=-sin, sin(t0+pi/2)=cos
    }
    float nv[3];
    nv[0] = a0[1]*a1[2] - a0[2]*a1[1];
    nv[1] = a0[2]*a1[0] - a0[0]*a1[2];
    nv[2] = a0[0]*a1[1] - a0[1]*a1[0];
    const float nn = sqrtf(nv[0]*nv[0] + nv[1]*nv[1] + nv[2]*nv[2]);
    const float ns = 0.001f / fmaxf(nn, 1e-12f);   // SIGMA = 0.001
    float WT[3][3];                                 // columns: [2*a0, 2*a1, n]
    #pragma unroll
    for (int k = 0; k < 3; ++k) {
        WT[k][0] = 2.0f * a0[k];
        WT[k][1] = 2.0f * a1[k];
        WT[k][2] = nv[k] * ns;
    }

    const float* wp = so3 + ((size_t)b * NF + f) * 3;
    const float wx = wp[0], wy = wp[1], wz = wp[2];
    const float th2 = wx*wx + wy*wy + wz*wz;
    const float t   = sqrtf(th2 + 1e-12f);
    const float A   = sinf(t) / t;
    const float Bc  = (1.0f - cosf(t)) / (t * t);
    float Rl[3][3];
    Rl[0][0] = 1.0f + Bc * (wx*wx - th2);
    Rl[0][1] = -A * wz + Bc * (wx*wy);
    Rl[0][2] =  A * wy + Bc * (wx*wz);
    Rl[1][0] =  A * wz + Bc * (wy*wx);
    Rl[1][1] = 1.0f + Bc * (wy*wy - th2);
    Rl[1][2] = -A * wx + Bc * (wy*wz);
    Rl[2][0] = -A * wy + Bc * (wz*wx);
    Rl[2][1] =  A * wx + Bc * (wz*wy);
    Rl[2][2] = 1.0f + Bc * (wz*wz - th2);

    const float* sp = scale + ((size_t)b * NF + f) * 3;
    const float s2x = sp[0]*sp[0], s2y = sp[1]*sp[1], s2z = sp[2]*sp[2];

    float CL[3][3];
    #pragma unroll
    for (int i = 0; i < 3; ++i)
        #pragma unroll
        for (int j = 0; j < 3; ++j)
            CL[i][j] = Rl[i][0]*s2x*Rl[j][0] + Rl[i][1]*s2y*Rl[j][1] + Rl[i][2]*s2z*Rl[j][2];

    float M1[3][3];
    #pragma unroll
    for (int i = 0; i < 3; ++i)
        #pragma unroll
        for (int j = 0; j < 3; ++j)
            M1[i][j] = WT[i][0]*CL[0][j] + WT[i][1]*CL[1][j] + WT[i][2]*CL[2][j];

    const float* op = offset + ((size_t)b * NF + f) * 3;
    float* dst = out + ((size_t)bn * NF + f) * 12;
    #pragma unroll
    for (int i = 0; i < 3; ++i) {
        dst[i * 4 + 0] = cen[i] + WT[i][0]*op[0] + WT[i][1]*op[1] + WT[i][2]*op[2];
        #pragma unroll
        for (int j = 0; j < 3; ++j)
            dst[i * 4 + 1 + j] = M1[i][0]*WT[j][0] + M1[i][1]*WT[j][1] + M1[i][2]*WT[j][2];
    }
}

// ---------------------------------------------------------------------------
extern "C" void kernel_launch(void* const* d_in, const int* in_sizes, int n_in,
                              void* d_out, int out_size, void* d_ws, size_t ws_size,
                              hipStream_t stream) {
    const float* xyz    = (const float*)d_in[0];
    const float* lbs    = (const float*)d_in[1];
    const float* Rg     = (const float*)d_in[2];
    const float* Tg     = (const float*)d_in[3];
    const float* iw     = (const float*)d_in[4];
    const float* so3    = (const float*)d_in[5];
    const float* scale  = (const float*)d_in[6];
    const float* offset = (const float*)d_in[7];
    const int*   ip     = (const int*)d_in[8];
    const int*   faces  = (const int*)d_in[9];
    float*       out    = (float*)d_out;

    const int NF  = in_sizes[9] / 3;
    const int NPH = in_sizes[8] / 3;
    const int B   = in_sizes[5] / (3 * NF);
    const int NP  = in_sizes[0] / (3 * B);
    const int J   = in_sizes[1] / (B * NP);
    const int N   = in_sizes[2] / (B * J * 9);
    const int BN  = B * N;

    float* posed = (float*)d_ws;                       // BN*NP*3 floats
    float* skel  = posed + (size_t)BN * NP * 3;        // BN*NPH*3 floats

    // Stage 1: WMMA skinning GEMM (one wave per 16-point tile)
    const int tiles = (NP + 15) / 16;
    if (J == 55) {
        skin_wmma_kernel<55><<<dim3(tiles, BN), 32, 0, stream>>>(xyz, lbs, Rg, Tg, posed, N, NP, J);
    } else {
        skin_wmma_kernel<0><<<dim3(tiles, BN), 32, 0, stream>>>(xyz, lbs, Rg, Tg, posed, N, NP, J);
    }

    // Stage 2: barycentric interpolation gather
    const int tot2 = BN * NPH;
    interp_kernel<<<(tot2 + 255) / 256, 256, 0, stream>>>(posed, iw, ip, skel, NP, NPH, tot2);

    // Stage 3: per-face Steiner / covariance
    const int tot3 = BN * NF;
    face_kernel<<<(tot3 + 255) / 256, 256, 0, stream>>>(skel, so3, scale, offset, faces, out,
                                                        N, NPH, NF, tot3);
}